// Model_71270687310453
// MI455X (gfx1250) — compile-verified
//
#include <hip/hip_runtime.h>

// ---------------- problem constants (from reference) ----------------
constexpr int BATCH = 16;
constexpr int LSEQ  = 512;
constexpr int NNODE = 1024;
constexpr int NEDGE = 16384;
constexpr int HDIM  = 256;
constexpr int NCLS  = 50;
constexpr int NREL1 = 51;             // R+1
constexpr int SEQ   = LSEQ + NCLS;    // 562
constexpr int NHEAD = 8;
constexpr int DHEAD = HDIM / NHEAD;   // 32
constexpr int FFD   = 4 * HDIM;       // 1024
constexpr int CLSD  = 1024;

typedef __attribute__((ext_vector_type(2))) float v2f;
typedef __attribute__((ext_vector_type(8))) float v8f;

// ---------------- generic f32 WMMA GEMM, 16(M) x 64(N) tile per wave ----------------
// C[b] = act( A[b] @ B[b] + bias ); A MxK row-major (optional row gather via idx),
// B KxN row-major.  N must be a multiple of 64 (true for all call sites: 256/1024).
// V_WMMA_F32_16X16X4_F32 layouts (ISA 7.12.2):
//   A 16x4 : lane&15 = M row; VGPR{0,1} = K {0,1} (lanes 0-15) / K {2,3} (lanes 16-31)
//   B 4x16 : lane&15 = N col; same K split across lane halves
//   C 16x16: n = lane&15, m = r + 8*(lane>=16) for VGPR r
// One A fragment feeds 4 WMMAs (4 adjacent N tiles) -> 4x less A traffic,
// 4x higher WMMA density per load batch.
__global__ __launch_bounds__(32) void wmma_gemm_kernel(
    const float* __restrict__ A, long strideA,
    const int* __restrict__ idx, long strideIdx,
    const float* __restrict__ Bm, long strideB,
    float* __restrict__ Cm, long strideC,
    int Mdim, int Ndim, int Kdim, int lda, int ldb, int ldc,
    const float* __restrict__ bias, int relu)
{
  const int lane = threadIdx.x;
  const int half = lane >> 4;           // 0 or 1
  const int l16  = lane & 15;
  const int bz   = blockIdx.z;
  const int m0   = blockIdx.y * 16;
  const int n0   = blockIdx.x * 64;

  const float* Ab = A  + (size_t)bz * strideA;
  const float* Bb = Bm + (size_t)bz * strideB;
  float*       Cb = Cm + (size_t)bz * strideC;

  const int  am    = m0 + l16;
  const bool am_ok = am < Mdim;
  size_t arow;
  if (idx) {
    const int* ib = idx + (size_t)bz * strideIdx;
    arow = (size_t)(am_ok ? ib[am] : 0) * (size_t)lda;
  } else {
    arow = (size_t)(am_ok ? am : 0) * (size_t)lda;
  }
  const int bn = n0 + l16;

  v8f acc0 = {}, acc1 = {}, acc2 = {}, acc3 = {};
  for (int k0 = 0; k0 < Kdim; k0 += 4) {
    const int ka = k0 + 2 * half;
    v2f av;
    const float a0 = Ab[arow + ka];
    const float a1 = Ab[arow + ka + 1];
    av.x = am_ok ? a0 : 0.0f;
    av.y = am_ok ? a1 : 0.0f;

    const float* Br0 = Bb + (size_t)ka * ldb + bn;
    const float* Br1 = Bb + (size_t)(ka + 1) * ldb + bn;
    v2f bv0, bv1, bv2, bv3;
    bv0.x = Br0[0];  bv0.y = Br1[0];
    bv1.x = Br0[16]; bv1.y = Br1[16];
    bv2.x = Br0[32]; bv2.y = Br1[32];
    bv3.x = Br0[48]; bv3.y = Br1[48];

    acc0 = __builtin_amdgcn_wmma_f32_16x16x4_f32(false, av, false, bv0, (short)0, acc0, false, false);
    acc1 = __builtin_amdgcn_wmma_f32_16x16x4_f32(false, av, false, bv1, (short)0, acc1, false, false);
    acc2 = __builtin_amdgcn_wmma_f32_16x16x4_f32(false, av, false, bv2, (short)0, acc2, false, false);
    acc3 = __builtin_amdgcn_wmma_f32_16x16x4_f32(false, av, false, bv3, (short)0, acc3, false, false);
  }

  const int mrow = m0 + 8 * half;
  #pragma unroll
  for (int j = 0; j < 4; ++j) {
    const v8f& acc = j == 0 ? acc0 : (j == 1 ? acc1 : (j == 2 ? acc2 : acc3));
    const int n = n0 + 16 * j + l16;
    float bsv = bias ? bias[n] : 0.0f;
    #pragma unroll
    for (int r = 0; r < 8; ++r) {
      const int m = mrow + r;
      if (m < Mdim) {
        float vvv = acc[r] + bsv;
        if (relu) vvv = vvv > 0.0f ? vvv : 0.0f;
        Cb[(size_t)m * ldc + n] = vvv;
      }
    }
  }
}

// ---------------- small helper kernels ----------------
__global__ void zero_kernel(float* p, long n) {
  long i = (long)blockIdx.x * blockDim.x + threadIdx.x;
  if (i < n) p[i] = 0.0f;
}

// edge_a[r] = dot(edge_emb[r,:], a_edge)
__global__ void edge_a_kernel(const float* __restrict__ edge_emb,
                              const float* __restrict__ a_e,
                              float* __restrict__ edge_a) {
  __shared__ float red[256];
  const int r = blockIdx.x, t = threadIdx.x;
  red[t] = edge_emb[(size_t)r * HDIM + t] * a_e[t];
  __syncthreads();
  for (int s = 128; s > 0; s >>= 1) {
    if (t < s) red[t] += red[t + s];
    __syncthreads();
  }
  if (t == 0) edge_a[r] = red[0];
}

// sal[i]=h[i,:]@a_src, dal[i]=h[i,:]@a_dst  over flattened (b,n)
__global__ void node_alpha_kernel(const float* __restrict__ h,
                                  const float* __restrict__ a_s,
                                  const float* __restrict__ a_d,
                                  float* __restrict__ sal,
                                  float* __restrict__ dal) {
  __shared__ float r1[256], r2[256];
  const int i = blockIdx.x, t = threadIdx.x;
  const float x = h[(size_t)i * HDIM + t];
  r1[t] = x * a_s[t];
  r2[t] = x * a_d[t];
  __syncthreads();
  for (int s = 128; s > 0; s >>= 1) {
    if (t < s) { r1[t] += r1[t + s]; r2[t] += r2[t + s]; }
    __syncthreads();
  }
  if (t == 0) { sal[i] = r1[0]; dal[i] = r2[0]; }
}

// monotonic float<->uint encoding for atomic max on floats
__device__ __forceinline__ unsigned fenc(float f) {
  unsigned u = __float_as_uint(f);
  return (u & 0x80000000u) ? ~u : (u | 0x80000000u);
}
__device__ __forceinline__ float fdec(unsigned u) {
  u = (u & 0x80000000u) ? (u & 0x7FFFFFFFu) : ~u;
  return __uint_as_float(u);
}

// e = leaky_relu(sal[src]+dal[dst]+edge_a LUT, 0.2); segment max via atomic umax
__global__ void edge_pass1(const int* __restrict__ edges,
                           const int* __restrict__ eattr,
                           const float* __restrict__ edge_a,
                           const float* __restrict__ sal,
                           const float* __restrict__ dal,
                           float* __restrict__ ev,
                           unsigned* __restrict__ segmax) {
  const int i = blockIdx.x * blockDim.x + threadIdx.x;
  if (i >= BATCH * NEDGE) return;
  const int b = i / NEDGE, ei = i - b * NEDGE;
  const int src = edges[(size_t)b * 2 * NEDGE + ei];
  const int dst = edges[(size_t)b * 2 * NEDGE + NEDGE + ei];
  const int attr = eattr[i];
  const float ee = edge_a[attr % NREL1] + edge_a[(attr / NREL1) % NREL1];
  float e = sal[b * NNODE + src] + dal[b * NNODE + dst] + ee;
  e = e > 0.0f ? e : 0.2f * e;
  ev[i] = e;
  atomicMax(&segmax[b * NNODE + dst], fenc(e));
}

// ex = exp(e - segmax[dst]); segment sum via atomicAdd
__global__ void edge_pass2(const int* __restrict__ edges,
                           const float* __restrict__ ev,
                           const unsigned* __restrict__ segmax,
                           float* __restrict__ ex,
                           float* __restrict__ den) {
  const int i = blockIdx.x * blockDim.x + threadIdx.x;
  if (i >= BATCH * NEDGE) return;
  const int b = i / NEDGE, ei = i - b * NEDGE;
  const int dst = edges[(size_t)b * 2 * NEDGE + NEDGE + ei];
  const float m = fdec(segmax[b * NNODE + dst]);
  const float e = __expf(ev[i] - m);
  ex[i] = e;
  atomicAdd(&den[b * NNODE + dst], e);
}

// agg[b,dst,:] += alpha * h[b,src,:]  (only dst<NCLS rows are ever consumed)
__global__ void edge_pass3(const int* __restrict__ edges,
                           const float* __restrict__ ex,
                           const float* __restrict__ den,
                           const float* __restrict__ h,
                           float* __restrict__ agg) {
  const int warp = threadIdx.x >> 5, lane = threadIdx.x & 31;
  const long i = (long)blockIdx.x * 8 + warp;
  if (i >= (long)BATCH * NEDGE) return;
  const int b = (int)(i / NEDGE), ei = (int)(i - (long)b * NEDGE);
  const int dst = edges[(size_t)b * 2 * NEDGE + NEDGE + ei];
  if (dst >= NCLS) return;
  const int src = edges[(size_t)b * 2 * NEDGE + ei];
  const float alpha = ex[i] / (den[b * NNODE + dst] + 1e-16f);
  const float* hrow = h + ((size_t)b * NNODE + src) * HDIM;
  float* arow = agg + ((size_t)b * NCLS + dst) * HDIM;
  for (int c = lane; c < HDIM; c += 32) atomicAdd(&arow[c], alpha * hrow[c]);
}

// merge = concat([word[:, :1], tanh(agg)+class_emb, word[:, 1:]])
__global__ void merge_kernel(const int* __restrict__ sentence,
                             const float* __restrict__ word_emb,
                             const float* __restrict__ agg,
                             const float* __restrict__ class_emb,
                             float* __restrict__ merge) {
  const int row = blockIdx.x, t = threadIdx.x;
  const int b = row / SEQ, s = row - b * SEQ;
  float v;
  if (s == 0) {
    v = word_emb[(size_t)sentence[b * LSEQ] * HDIM + t];
  } else if (s <= NCLS) {
    v = tanhf(agg[((size_t)b * NCLS + (s - 1)) * HDIM + t]) +
        class_emb[(size_t)(s - 1) * HDIM + t];
  } else {
    v = word_emb[(size_t)sentence[b * LSEQ + (s - NCLS)] * HDIM + t];
  }
  merge[(size_t)row * HDIM + t] = v;
}

// per-(b,head,query-row) softmax attention; dh=32, S=562
__global__ void attn_kernel(const float* __restrict__ q,
                            const float* __restrict__ k,
                            const float* __restrict__ v,
                            const float* __restrict__ mask,
                            float* __restrict__ att) {
  const int qi = blockIdx.x, hd = blockIdx.y, b = blockIdx.z;
  const int tid = threadIdx.x;
  __shared__ float sc[SEQ];
  __shared__ float red[128];
  __shared__ float qv[DHEAD];

  const float* qrow = q + ((size_t)(b * SEQ + qi)) * HDIM + hd * DHEAD;
  if (tid < DHEAD) qv[tid] = qrow[tid];
  __syncthreads();

  const float scale = 0.17677669529663687f;  // 1/sqrt(32)
  for (int j = tid; j < SEQ; j += 128) {
    const float* krow = k + ((size_t)(b * SEQ + j)) * HDIM + hd * DHEAD;
    float d = 0.0f;
    #pragma unroll
    for (int t = 0; t < DHEAD; ++t) d += qv[t] * krow[t];
    sc[j] = d * scale + (mask[b * SEQ + j] - 1.0f) * 1e9f;
  }
  __syncthreads();

  float mx = -1e30f;
  for (int j = tid; j < SEQ; j += 128) mx = fmaxf(mx, sc[j]);
  red[tid] = mx; __syncthreads();
  for (int s = 64; s > 0; s >>= 1) {
    if (tid < s) red[tid] = fmaxf(red[tid], red[tid + s]);
    __syncthreads();
  }
  mx = red[0]; __syncthreads();

  float sm = 0.0f;
  for (int j = tid; j < SEQ; j += 128) {
    const float e = __expf(sc[j] - mx);
    sc[j] = e;
    sm += e;
  }
  red[tid] = sm; __syncthreads();
  for (int s = 64; s > 0; s >>= 1) {
    if (tid < s) red[tid] += red[tid + s];
    __syncthreads();
  }
  const float inv = 1.0f / red[0];
  __syncthreads();

  // output: d = tid&31, 4-way split of the j loop
  const int d = tid & 31, part = tid >> 5;
  float acc = 0.0f;
  for (int j = part; j < SEQ; j += 4)
    acc += sc[j] * v[((size_t)(b * SEQ + j)) * HDIM + hd * DHEAD + d];
  red[tid] = acc; __syncthreads();
  if (tid < 32) {
    const float o = (red[tid] + red[tid + 32] + red[tid + 64] + red[tid + 96]) * inv;
    att[((size_t)(b * SEQ + qi)) * HDIM + hd * DHEAD + d] = o;
  }
}

// out = LayerNorm(a + bb) * g + beta  (H=256, one row per block; out may alias a)
__global__ void add_ln_kernel(const float* a, const float* bb,
                              const float* __restrict__ g,
                              const float* __restrict__ beta,
                              float* out) {
  __shared__ float red[256];
  __shared__ float s_mean, s_inv;
  const int row = blockIdx.x, t = threadIdx.x;
  const size_t base = (size_t)row * HDIM;
  const float x = a[base + t] + bb[base + t];
  red[t] = x; __syncthreads();
  for (int s = 128; s > 0; s >>= 1) {
    if (t < s) red[t] += red[t + s];
    __syncthreads();
  }
  if (t == 0) s_mean = red[0] * (1.0f / HDIM);
  __syncthreads();
  const float dv = x - s_mean;
  red[t] = dv * dv; __syncthreads();
  for (int s = 128; s > 0; s >>= 1) {
    if (t < s) red[t] += red[t + s];
    __syncthreads();
  }
  if (t == 0) s_inv = rsqrtf(red[0] * (1.0f / HDIM) + 1e-5f);
  __syncthreads();
  out[base + t] = dv * s_inv * g[t] + beta[t];
}

// y[b,c] = sum_k cls_W2[c,k] * hid[b,c,k]
__global__ void cls_out_kernel(const float* __restrict__ hid,
                               const float* __restrict__ W2,
                               float* __restrict__ y) {
  __shared__ float red[256];
  const int i = blockIdx.x, t = threadIdx.x;
  const int c = i % NCLS;
  const float* hrow = hid + (size_t)i * CLSD;
  const float* wrow = W2 + (size_t)c * CLSD;
  float s = 0.0f;
  for (int k = t; k < CLSD; k += 256) s += hrow[k] * wrow[k];
  red[t] = s; __syncthreads();
  for (int s2 = 128; s2 > 0; s2 >>= 1) {
    if (t < s2) red[t] += red[t + s2];
    __syncthreads();
  }
  if (t == 0) y[i] = red[0];
}

// ---------------- orchestration ----------------
extern "C" void kernel_launch(void* const* d_in, const int* in_sizes, int n_in,
                              void* d_out, int out_size, void* d_ws, size_t ws_size,
                              hipStream_t stream) {
  (void)in_sizes; (void)n_in; (void)out_size; (void)ws_size;
  const int*   sentence  = (const int*)  d_in[0];
  const float* mask      = (const float*)d_in[1];
  const int*   nodes     = (const int*)  d_in[2];
  const int*   edges     = (const int*)  d_in[3];
  const int*   edge_attr = (const int*)  d_in[4];
  const float* word_emb  = (const float*)d_in[5];
  const float* node_emb  = (const float*)d_in[6];
  const float* edge_emb  = (const float*)d_in[7];
  const float* class_emb = (const float*)d_in[8];
  const float* gat_W     = (const float*)d_in[9];
  const float* a_src     = (const float*)d_in[10];
  const float* a_dst     = (const float*)d_in[11];
  const float* a_edge    = (const float*)d_in[12];
  const float* Wq        = (const float*)d_in[13];
  const float* Wk        = (const float*)d_in[14];
  const float* Wv        = (const float*)d_in[15];
  const float* Wo        = (const float*)d_in[16];
  const float* ln1_g     = (const float*)d_in[17];
  const float* ln1_b     = (const float*)d_in[18];
  const float* ff_W1     = (const float*)d_in[19];
  const float* ff_b1     = (const float*)d_in[20];
  const float* ff_W2     = (const float*)d_in[21];
  const float* ff_b2     = (const float*)d_in[22];
  const float* ln2_g     = (const float*)d_in[23];
  const float* ln2_b     = (const float*)d_in[24];
  const float* cls_W1    = (const float*)d_in[25];
  const float* cls_b1    = (const float*)d_in[26];
  const float* cls_W2    = (const float*)d_in[27];
  float* ws = (float*)d_ws;
  float* y  = (float*)d_out;

  // workspace arena (floats); segmax/den/agg kept contiguous for one zero pass
  size_t off = 0;
  float* h      = ws + off; off += (size_t)BATCH * NNODE * HDIM;   // 16 MB
  float* sal    = ws + off; off += (size_t)BATCH * NNODE;
  float* dal    = ws + off; off += (size_t)BATCH * NNODE;
  float* segmax = ws + off; off += (size_t)BATCH * NNODE;          // encoded uint
  float* den    = ws + off; off += (size_t)BATCH * NNODE;
  float* agg    = ws + off; off += (size_t)BATCH * NCLS * HDIM;
  float* edge_a = ws + off; off += 64;
  float* ev     = ws + off; off += (size_t)BATCH * NEDGE;
  float* exb    = ws + off; off += (size_t)BATCH * NEDGE;
  float* merge  = ws + off; off += (size_t)BATCH * SEQ * HDIM;     // also x after LN1
  float* qb     = ws + off; off += (size_t)BATCH * SEQ * HDIM;     // reused: ff2 out
  float* kb     = ws + off; off += (size_t)BATCH * SEQ * HDIM;     // reused: text_vec
  float* vb     = ws + off; off += (size_t)BATCH * SEQ * HDIM;     // reused: cls hidden
  float* att    = ws + off; off += (size_t)BATCH * SEQ * HDIM;
  float* oproj  = ws + off; off += (size_t)BATCH * SEQ * HDIM;
  float* ffh    = ws + off; off += (size_t)BATCH * SEQ * FFD;      // 37 MB

  const dim3 wave(32);
  const int EALL = BATCH * NEDGE;

  // --- GAT ---
  // h[b] = node_emb[nodes[b]] @ gat_W   (fused gather + WMMA GEMM)
  wmma_gemm_kernel<<<dim3(HDIM / 64, NNODE / 16, BATCH), wave, 0, stream>>>(
      node_emb, 0, nodes, NNODE, gat_W, 0, h, (long)NNODE * HDIM,
      NNODE, HDIM, HDIM, HDIM, HDIM, HDIM, nullptr, 0);

  edge_a_kernel<<<NREL1, 256, 0, stream>>>(edge_emb, a_edge, edge_a);
  node_alpha_kernel<<<BATCH * NNODE, 256, 0, stream>>>(h, a_src, a_dst, sal, dal);

  const long zcount = (long)BATCH * NNODE * 2 + (long)BATCH * NCLS * HDIM;
  zero_kernel<<<(int)((zcount + 255) / 256), 256, 0, stream>>>(segmax, zcount);

  edge_pass1<<<(EALL + 255) / 256, 256, 0, stream>>>(
      edges, edge_attr, edge_a, sal, dal, ev, (unsigned*)segmax);
  edge_pass2<<<(EALL + 255) / 256, 256, 0, stream>>>(
      edges, ev, (const unsigned*)segmax, exb, den);
  edge_pass3<<<(EALL + 7) / 8, 256, 0, stream>>>(edges, exb, den, h, agg);

  // --- merge token stream ---
  merge_kernel<<<BATCH * SEQ, 256, 0, stream>>>(sentence, word_emb, agg, class_emb, merge);

  // --- encoder ---
  const dim3 gH(HDIM / 64, (SEQ + 15) / 16, BATCH);
  wmma_gemm_kernel<<<gH, wave, 0, stream>>>(merge, (long)SEQ * HDIM, nullptr, 0,
      Wq, 0, qb, (long)SEQ * HDIM, SEQ, HDIM, HDIM, HDIM, HDIM, HDIM, nullptr, 0);
  wmma_gemm_kernel<<<gH, wave, 0, stream>>>(merge, (long)SEQ * HDIM, nullptr, 0,
      Wk, 0, kb, (long)SEQ * HDIM, SEQ, HDIM, HDIM, HDIM, HDIM, HDIM, nullptr, 0);
  wmma_gemm_kernel<<<gH, wave, 0, stream>>>(merge, (long)SEQ * HDIM, nullptr, 0,
      Wv, 0, vb, (long)SEQ * HDIM, SEQ, HDIM, HDIM, HDIM, HDIM, HDIM, nullptr, 0);

  attn_kernel<<<dim3(SEQ, NHEAD, BATCH), 128, 0, stream>>>(qb, kb, vb, mask, att);

  wmma_gemm_kernel<<<gH, wave, 0, stream>>>(att, (long)SEQ * HDIM, nullptr, 0,
      Wo, 0, oproj, (long)SEQ * HDIM, SEQ, HDIM, HDIM, HDIM, HDIM, HDIM, nullptr, 0);

  add_ln_kernel<<<BATCH * SEQ, 256, 0, stream>>>(merge, oproj, ln1_g, ln1_b, merge);

  wmma_gemm_kernel<<<dim3(FFD / 64, (SEQ + 15) / 16, BATCH), wave, 0, stream>>>(
      merge, (long)SEQ * HDIM, nullptr, 0, ff_W1, 0, ffh, (long)SEQ * FFD,
      SEQ, FFD, HDIM, HDIM, FFD, FFD, ff_b1, 1);
  wmma_gemm_kernel<<<gH, wave, 0, stream>>>(
      ffh, (long)SEQ * FFD, nullptr, 0, ff_W2, 0, qb, (long)SEQ * HDIM,
      SEQ, HDIM, FFD, FFD, HDIM, HDIM, ff_b2, 0);

  add_ln_kernel<<<BATCH * SEQ, 256, 0, stream>>>(merge, qb, ln2_g, ln2_b, kb);

  // --- classifier: only rows 1..NCLS of text_vec are consumed ---
  wmma_gemm_kernel<<<dim3(CLSD / 64, (NCLS + 15) / 16, BATCH), wave, 0, stream>>>(
      kb + HDIM, (long)SEQ * HDIM, nullptr, 0, cls_W1, 0, vb, (long)NCLS * CLSD,
      NCLS, CLSD, HDIM, HDIM, CLSD, CLSD, cls_b1, 1);

  cls_out_kernel<<<BATCH * NCLS, 256, 0, stream>>>(vb, cls_W2, y);
}